// Closing2d_24120536334522
// MI455X (gfx1250) — compile-verified
//
#include <hip/hip_runtime.h>
#include <stdint.h>

#define HW 256
#define PADV 10000.0f
#define RB 32   // output rows per block (band height)

typedef __attribute__((ext_vector_type(4))) unsigned int u32x4;
typedef __attribute__((ext_vector_type(8))) unsigned int u32x8;

// Fused separable weighted dilation/erosion over one band of RB rows.
//   IS_MAX=1 : out = max_j ( src[.+j] - c*se[j] ), pad -PADV  (dilation)
//   IS_MAX=0 : out = min_j ( src[.+j] + c*se[j] ), pad +PADV  (erosion)
// LDS tile: Rt=(RB+2W) rows x P=(256+2W) cols, data at col offset W.
// TDM loads the valid rows; its pad feature (interval=256 dwords,
// amount=2W dwords) strides rows directly into the padded layout.
template <int W, int IS_MAX>
__global__ void __launch_bounds__(256)
morph_pass(const float* __restrict__ src, float* __restrict__ dst,
           const float* __restrict__ se_coef_p, float inv4t,
           long long dstImgStride, long long dstBase)
{
    constexpr int P  = HW + 2 * W;   // padded LDS row stride (floats)
    constexpr int Rt = RB + 2 * W;   // rows resident in LDS
    constexpr int N  = 2 * W + 1;    // SE length
    extern __shared__ float smem[];

    const int bc  = blockIdx.y;       // image index (B*C = 32)
    const int r0  = blockIdx.x * RB;  // first output row of band
    const int tid = threadIdx.x;
    const int g   = tid >> 6;         // row group 0..3
    const int c4  = (tid & 63) << 2;  // first of this thread's 4 columns

    const float cs   = se_coef_p[0] * inv4t;       // c / (4*t_k)
    const float padv = IS_MAX ? -PADV : PADV;

    // ---- 1) pre-fill tile with pad value (pad cols + out-of-image rows)
    {
        float4  pv = make_float4(padv, padv, padv, padv);
        float4* s4 = (float4*)smem;
        constexpr int N4 = (Rt * P) >> 2;  // Rt*P divisible by 4
        #pragma unroll 2
        for (int i = tid; i < N4; i += 256) s4[i] = pv;
    }
    __syncthreads();

    // ---- 2) TDM: DMA valid rows of the band into the tile interior
    {
        int gFirst = r0 - W, loFirst = 0;
        if (gFirst < 0) { loFirst = -gFirst; gFirst = 0; }
        int gLast = r0 + RB + W; if (gLast > HW) gLast = HW;
        const int nRows = gLast - gFirst;
        if (tid == 0) {
            unsigned lds_byte = (unsigned)(size_t)(smem + (size_t)loFirst * P + W);
            unsigned long long ga =
                (unsigned long long)(size_t)(src + (size_t)bc * HW * HW
                                                 + (size_t)gFirst * HW);
            u32x4 g0; u32x8 g1;
            g0[0] = 1u;                                   // count=1
            g0[1] = lds_byte;                             // lds_addr
            g0[2] = (unsigned)(ga & 0xFFFFFFFFu);         // global_addr lo
            g0[3] = (unsigned)((ga >> 32) & 0x1FFFFFFu) | (2u << 30); // hi|type
            constexpr unsigned padAmt = (unsigned)(2 * W - 1); // 2W dwords
            g1[0] = (2u << 16) | (1u << 20) | (7u << 22) | (padAmt << 25);
            g1[1] = ((unsigned)HW & 0xFFFFu) << 16;   // tensor_dim0 lo16
            g1[2] = ((unsigned)nRows & 0xFFFFu) << 16;// dim0 hi=0 | dim1 lo16
            g1[3] = ((unsigned)HW & 0xFFFFu) << 16;   // dim1 hi=0 | tile_dim0
            g1[4] = ((unsigned)nRows & 0xFFFFu);      // tile_dim1 | tile_dim2=0
            g1[5] = (unsigned)HW;                     // tensor_dim0_stride
            g1[6] = 0u; g1[7] = 0u;
            asm volatile("tensor_load_to_lds %0, %1" :: "s"(g0), "s"(g1) : "memory");
            __builtin_amdgcn_s_wait_tensorcnt(0);
        }
    }
    __syncthreads();

    // ---- 3) horizontal pass, in-place: 4 rows/sweep, 4 cols/thread,
    //         16B-aligned ds_load_b128 window chunks, immediate offsets.
    for (int li = 0; li < Rt; li += 4) {
        const float* rowp = smem + (size_t)(li + g) * P + c4; // padded col 4c
        float a0, a1, a2, a3;
        a0 = a1 = a2 = a3 = padv;
        constexpr int NC = W / 2 + 1;   // chunks covering [4c, 4c+2W+3]
        #pragma unroll
        for (int m = 0; m < NC; ++m) {
            const float4 v = *(const float4*)(rowp + 4 * m);
            const float vv[4] = {v.x, v.y, v.z, v.w};
            #pragma unroll
            for (int i = 0; i < 4; ++i) {
                #pragma unroll
                for (int k = 0; k < 4; ++k) {
                    constexpr int dummy = 0; (void)dummy;
                    const int j = 4 * m + i - k;       // compile-time
                    if (j >= 0 && j <= 2 * W) {
                        const float e = cs * (float)((j - W) * (j - W));
                        float& a = (k == 0) ? a0 : (k == 1) ? a1
                                 : (k == 2) ? a2 : a3;
                        a = IS_MAX ? fmaxf(a, vv[i] - e)
                                   : fminf(a, vv[i] + e);
                    }
                }
            }
        }
        __syncthreads();                 // all reads done before overwrite
        *(float4*)(smem + (size_t)(li + g) * P + W + c4) =
            make_float4(a0, a1, a2, a3);
        __syncthreads();
    }

    // ---- 4) vertical pass: 4 output rows/sweep, 4 cols/thread,
    //         b128 column loads, weight e[j] shared across the 4 cols.
    const size_t dOff = (size_t)bc * (size_t)dstImgStride + (size_t)dstBase;
    for (int rr = 0; rr < RB; rr += 4) {
        const int orow = rr + g;                       // band-local out row
        const float* base = smem + (size_t)orow * P + W + c4;
        float a0, a1, a2, a3;
        a0 = a1 = a2 = a3 = padv;
        #pragma unroll
        for (int j = 0; j < N; ++j) {
            const float4 v = *(const float4*)(base + (size_t)j * P);
            const float e = cs * (float)((j - W) * (j - W));
            if (IS_MAX) {
                a0 = fmaxf(a0, v.x - e); a1 = fmaxf(a1, v.y - e);
                a2 = fmaxf(a2, v.z - e); a3 = fmaxf(a3, v.w - e);
            } else {
                a0 = fminf(a0, v.x + e); a1 = fminf(a1, v.y + e);
                a2 = fminf(a2, v.z + e); a3 = fminf(a3, v.w + e);
            }
        }
        *(float4*)(dst + dOff + (size_t)(r0 + orow) * HW + c4) =
            make_float4(a0, a1, a2, a3);
    }
}

template <int W>
static void launch_scale(const float* in, float* ws, float* out,
                         const float* coef, int s, hipStream_t stream)
{
    constexpr int Rt = RB + 2 * W;
    constexpr int P  = HW + 2 * W;
    const size_t lds = (size_t)Rt * P * sizeof(float);
    const float inv4t = 1.0f / (4.0f * (float)(1 << (2 * s))); // 1/(4*4^s)
    dim3 grid(HW / RB, 32), block(256);
    // dilation: in -> ws
    morph_pass<W, 1><<<grid, block, lds, stream>>>(
        in, ws, coef, inv4t, (long long)(HW * HW), (long long)0);
    // erosion: ws -> out[:, :, s, :, :]
    morph_pass<W, 0><<<grid, block, lds, stream>>>(
        ws, out, coef, inv4t, (long long)(4 * HW * HW),
        (long long)s * HW * HW);
}

extern "C" void kernel_launch(void* const* d_in, const int* in_sizes, int n_in,
                              void* d_out, int out_size, void* d_ws, size_t ws_size,
                              hipStream_t stream) {
    (void)in_sizes; (void)n_in; (void)out_size; (void)ws_size;
    const float* in   = (const float*)d_in[0];   // [4,8,256,256] f32
    const float* coef = (const float*)d_in[1];   // scalar se_coef (device)
    float*       out  = (float*)d_out;           // [4,8,4,256,256] f32
    float*       ws   = (float*)d_ws;            // 8 MB ping buffer

    launch_scale<4>(in, ws, out, coef, 0, stream);
    launch_scale<8>(in, ws, out, coef, 1, stream);
    launch_scale<16>(in, ws, out, coef, 2, stream);
    launch_scale<32>(in, ws, out, coef, 3, stream);
}